// NTMReadController_1477468749968
// MI455X (gfx1250) — compile-verified
//
#include <hip/hip_runtime.h>
#include <hip/hip_bf16.h>
#include <math.h>

typedef float v2f __attribute__((ext_vector_type(2)));
typedef float v8f __attribute__((ext_vector_type(8)));

#define EPSF 1e-8f
#define NB 64
#define NN 65536
#define MD 64
#define DD 256
#define AC 70          // M + 6
#define CHUNKS 256     // n-chunks per batch (256 n each)

__device__ __forceinline__ float softplusf(float x) {
    return (x > 20.0f) ? x : log1pf(expf(x));
}

// ---------------------------------------------------------------------------
// Kernel 1: a = emb(64x256) @ W(256x70) + b  via v_wmma_f32_16x16x4_f32.
// One wave per 16x16 C tile; 4 row-tiles x 5 col-tiles (cols padded to 80).
// A frag (16x4 f32): lanes 0-15 row M, v[0]=K0,v[1]=K1; lanes 16-31 K2,K3.
// B frag (4x16 f32): mirrored (lane = N col, halves pick K pairs).
// D (16x16 f32, 8 VGPRs): lane<16 -> rows 0-7, lane>=16 -> rows 8-15.
// Padding lanes use a clamped column + 0/1 mask so the inner loop has NO
// divergence (EXEC stays all-1s through the WMMA stream, per ISA rules).
// ---------------------------------------------------------------------------
__global__ __launch_bounds__(32) void ntm_gemm_wmma(
    const float* __restrict__ emb, const float* __restrict__ W,
    const float* __restrict__ bias, float* __restrict__ a_out)
{
    const int tile = blockIdx.x;        // 0..19
    const int tm = tile / 5;            // row tile
    const int tn = tile % 5;            // col tile
    const int lane = threadIdx.x;
    const int half = lane >> 4;         // 0/1 -> K sub-pair
    const int l = lane & 15;

    const int mrow = tm * 16 + l;              // A row served by this lane
    const int ncol = tn * 16 + l;              // B col served by this lane
    const int ncc  = (ncol < AC) ? ncol : (AC - 1);   // clamped (always in-range)
    const float msk = (ncol < AC) ? 1.0f : 0.0f;      // zero out padding lanes

    const float* __restrict__ arow = emb + mrow * DD;
    const float* __restrict__ wcol = W + ncc;

    v8f c = {};
    for (int k = 0; k < DD; k += 4) {
        const int k0 = k + half * 2;
        v2f af, bf;
        af[0] = arow[k0];
        af[1] = arow[k0 + 1];
        bf[0] = msk * wcol[(size_t)k0 * AC];
        bf[1] = msk * wcol[(size_t)(k0 + 1) * AC];
        c = __builtin_amdgcn_wmma_f32_16x16x4_f32(false, af, false, bf,
                                                  (short)0, c, false, false);
    }
    if (ncol < AC) {
        const float bv = bias[ncol];
        #pragma unroll
        for (int r = 0; r < 8; ++r) {
            const int m = tm * 16 + r + half * 8;
            a_out[m * AC + ncol] = c[r] + bv;
        }
    }
}

// ---------------------------------------------------------------------------
// Kernel 2: per-batch key normalization + gate scalars.
// sc layout per b (stride 8): beta, g, s0, s1, s2, gamma
// ---------------------------------------------------------------------------
__global__ void ntm_prep(const float* __restrict__ a, float* __restrict__ kn,
                         float* __restrict__ sc)
{
    const int b = blockIdx.x, t = threadIdx.x;   // 64 threads
    __shared__ float red[64];
    const float kv = a[b * AC + t];
    red[t] = kv * kv;
    __syncthreads();
    for (int s = 32; s > 0; s >>= 1) { if (t < s) red[t] += red[t + s]; __syncthreads(); }
    const float nrm = sqrtf(red[0]);
    kn[b * MD + t] = kv / (nrm + EPSF);
    if (t == 0) {
        const float beta  = softplusf(a[b * AC + 64]);
        const float g     = 1.0f / (1.0f + expf(-a[b * AC + 65]));
        const float s0 = a[b * AC + 66], s1 = a[b * AC + 67], s2 = a[b * AC + 68];
        const float mx = fmaxf(s0, fmaxf(s1, s2));
        const float e0 = expf(s0 - mx), e1 = expf(s1 - mx), e2 = expf(s2 - mx);
        const float se = e0 + e1 + e2;
        const float gamma = 1.0f + softplusf(a[b * AC + 69]);
        float* s = &sc[b * 8];
        s[0] = beta; s[1] = g; s[2] = e0 / se; s[3] = e1 / se; s[4] = e2 / se; s[5] = gamma;
    }
}

// ---------------------------------------------------------------------------
// Kernel 3: sim[b,n] = kn . mem[b,n,:] / (||mem[b,n,:]|| + eps)
// Streaming pass #1 over 1.07 GB. 256 threads: 16 rows x 16 float4-quads per
// step, 16 steps -> 256 rows/block. Coalesced 128-bit loads; shfl reductions.
// Also emits per-block max(sim).
// ---------------------------------------------------------------------------
__global__ __launch_bounds__(256) void ntm_sim(
    const float* __restrict__ mem, const float* __restrict__ kn,
    float* __restrict__ sim, float* __restrict__ bmax)
{
    const int b = blockIdx.y, chunk = blockIdx.x, t = threadIdx.x;
    const int q = t & 15;        // quad (m-range q*4..q*4+3)
    const int rloc = t >> 4;     // row within step

    __shared__ float skn[MD];
    __shared__ float ssim[256];
    if (t < MD) skn[t] = kn[b * MD + t];
    __syncthreads();
    const float4 kq = *(const float4*)&skn[q * 4];

    const size_t base = ((size_t)b * NN + (size_t)chunk * 256) * MD;
    for (int step = 0; step < 16; ++step) {
        const int n = step * 16 + rloc;
        const float4 v = *(const float4*)(mem + base + (size_t)n * MD + q * 4);
        float dot = kq.x * v.x + kq.y * v.y + kq.z * v.z + kq.w * v.w;
        float sq  = v.x * v.x + v.y * v.y + v.z * v.z + v.w * v.w;
        #pragma unroll
        for (int m2 = 8; m2 >= 1; m2 >>= 1) {
            dot += __shfl_xor(dot, m2, 32);
            sq  += __shfl_xor(sq,  m2, 32);
        }
        if (q == 0) ssim[n] = dot / (sqrtf(sq) + EPSF);
    }
    __syncthreads();
    sim[(size_t)b * NN + (size_t)chunk * 256 + t] = ssim[t];   // coalesced store
    __syncthreads();
    for (int s = 128; s > 0; s >>= 1) {                        // in-place max
        if (t < s) ssim[t] = fmaxf(ssim[t], ssim[t + s]);
        __syncthreads();
    }
    if (t == 0) bmax[b * CHUNKS + chunk] = ssim[0];
}

__global__ void ntm_reduce_max(const float* __restrict__ bp, float* __restrict__ g)
{
    const int b = blockIdx.x, t = threadIdx.x;   // 256 threads
    __shared__ float r[256];
    r[t] = bp[b * CHUNKS + t];
    __syncthreads();
    for (int s = 128; s > 0; s >>= 1) { if (t < s) r[t] = fmaxf(r[t], r[t + s]); __syncthreads(); }
    if (t == 0) g[b] = r[0];
}

__global__ void ntm_reduce_sum(const float* __restrict__ bp, float* __restrict__ g)
{
    const int b = blockIdx.x, t = threadIdx.x;   // 256 threads
    __shared__ float r[256];
    r[t] = bp[b * CHUNKS + t];
    __syncthreads();
    for (int s = 128; s > 0; s >>= 1) { if (t < s) r[t] += r[t + s]; __syncthreads(); }
    if (t == 0) g[b] = r[0];
}

// Kernel 5: e = exp(beta*(sim - gmax)) in place (beta>0 so max commutes) + block sums.
__global__ __launch_bounds__(256) void ntm_exp(
    float* __restrict__ e, const float* __restrict__ gmax,
    const float* __restrict__ sc, float* __restrict__ bsum)
{
    const int b = blockIdx.y, chunk = blockIdx.x, t = threadIdx.x;
    const size_t i = (size_t)b * NN + (size_t)chunk * 256 + t;
    const float v = expf(sc[b * 8 + 0] * (e[i] - gmax[b]));
    e[i] = v;
    __shared__ float r[256];
    r[t] = v;
    __syncthreads();
    for (int s = 128; s > 0; s >>= 1) { if (t < s) r[t] += r[t + s]; __syncthreads(); }
    if (t == 0) bsum[b * CHUNKS + chunk] = r[0];
}

// Kernel 7: wg(n)=g*e[n]/S+(1-g)*w_prev[n]; wt = s0*wg(n-1)+s1*wg(n)+s2*wg(n+1)
// (circular); wp = wt^gamma; + block sums of wp.
__global__ __launch_bounds__(256) void ntm_shift_pow(
    const float* __restrict__ e, const float* __restrict__ wprev,
    const float* __restrict__ sc, const float* __restrict__ gsum,
    float* __restrict__ wp, float* __restrict__ bsum)
{
    const int b = blockIdx.y, chunk = blockIdx.x, t = threadIdx.x;
    const int n = chunk * 256 + t;
    const float invS = 1.0f / gsum[b];
    const float g  = sc[b * 8 + 1];
    const float s0 = sc[b * 8 + 2], s1 = sc[b * 8 + 3], s2 = sc[b * 8 + 4];
    const float gamma = sc[b * 8 + 5];
    const int nm1 = (n == 0) ? (NN - 1) : (n - 1);
    const int np1 = (n == NN - 1) ? 0 : (n + 1);
    const size_t base = (size_t)b * NN;
    const float wg_m = g * e[base + nm1] * invS + (1.0f - g) * wprev[base + nm1];
    const float wg_0 = g * e[base + n  ] * invS + (1.0f - g) * wprev[base + n  ];
    const float wg_p = g * e[base + np1] * invS + (1.0f - g) * wprev[base + np1];
    const float wt = s0 * wg_m + s1 * wg_0 + s2 * wg_p;   // >= 0
    const float v = powf(wt, gamma);
    wp[base + n] = v;
    __shared__ float r[256];
    r[t] = v;
    __syncthreads();
    for (int s = 128; s > 0; s >>= 1) { if (t < s) r[t] += r[t + s]; __syncthreads(); }
    if (t == 0) bsum[b * CHUNKS + chunk] = r[0];
}

// Kernel 9: w = wp / (sum + eps) -> d_out (w section).
__global__ __launch_bounds__(256) void ntm_norm_w(
    const float* __restrict__ wp, const float* __restrict__ gsum2,
    float* __restrict__ wout)
{
    const int b = blockIdx.y, chunk = blockIdx.x, t = threadIdx.x;
    const size_t i = (size_t)b * NN + (size_t)chunk * 256 + t;
    wout[i] = wp[i] / (gsum2[b] + EPSF);
}

// ---------------------------------------------------------------------------
// Kernel 10: streaming pass #2: partial read[b,m] = sum_n w[b,n]*mem[b,n,m]
// Same coalesced float4 layout; shfl_xor(16) folds row pairs inside a wave,
// shared memory folds the 8 waves -> 64 partials per block.
// ---------------------------------------------------------------------------
__global__ __launch_bounds__(256) void ntm_read(
    const float* __restrict__ mem, const float* __restrict__ w,
    float* __restrict__ rpart)
{
    const int b = blockIdx.y, chunk = blockIdx.x, t = threadIdx.x;
    const int q = t & 15, rloc = t >> 4;
    const size_t base = ((size_t)b * NN + (size_t)chunk * 256) * MD;
    const float* wb = w + (size_t)b * NN + (size_t)chunk * 256;

    float4 acc = make_float4(0.f, 0.f, 0.f, 0.f);
    for (int step = 0; step < 16; ++step) {
        const int n = step * 16 + rloc;
        const float wv = wb[n];
        const float4 v = *(const float4*)(mem + base + (size_t)n * MD + q * 4);
        acc.x += wv * v.x; acc.y += wv * v.y; acc.z += wv * v.z; acc.w += wv * v.w;
    }
    acc.x += __shfl_xor(acc.x, 16, 32);
    acc.y += __shfl_xor(acc.y, 16, 32);
    acc.z += __shfl_xor(acc.z, 16, 32);
    acc.w += __shfl_xor(acc.w, 16, 32);

    __shared__ float sm[8][64];
    const int wave = t >> 5, lane = t & 31;
    if (lane < 16) {
        sm[wave][q * 4 + 0] = acc.x; sm[wave][q * 4 + 1] = acc.y;
        sm[wave][q * 4 + 2] = acc.z; sm[wave][q * 4 + 3] = acc.w;
    }
    __syncthreads();
    if (t < MD) {
        float s = 0.f;
        #pragma unroll
        for (int wv2 = 0; wv2 < 8; ++wv2) s += sm[wv2][t];
        rpart[(((size_t)b * CHUNKS) + chunk) * MD + t] = s;
    }
}

// Kernel 11: fold chunk partials -> read[b,m] into d_out[0:4096).
__global__ void ntm_read_reduce(const float* __restrict__ rpart, float* __restrict__ out)
{
    const int b = blockIdx.x, m = threadIdx.x;   // 64 threads
    float s = 0.f;
    for (int c = 0; c < CHUNKS; ++c)
        s += rpart[(((size_t)b * CHUNKS) + c) * MD + m];
    out[b * MD + m] = s;
}

// ---------------------------------------------------------------------------
extern "C" void kernel_launch(void* const* d_in, const int* in_sizes, int n_in,
                              void* d_out, int out_size, void* d_ws, size_t ws_size,
                              hipStream_t stream)
{
    const float* emb   = (const float*)d_in[0];   // 64*256
    const float* wprev = (const float*)d_in[1];   // 64*65536
    const float* mem   = (const float*)d_in[2];   // 64*65536*64
    const float* W     = (const float*)d_in[3];   // 256*70
    const float* bias  = (const float*)d_in[4];   // 70
    float* out = (float*)d_out;                   // [read 64*64][w 64*65536]
    float* ws  = (float*)d_ws;

    // workspace layout (float offsets)
    const size_t A_OFF     = 0;                       // 64*70
    const size_t KN_OFF    = 5120;                    // 64*64
    const size_t SC_OFF    = 9216;                    // 64*8
    const size_t GMAX_OFF  = 10240;                   // 64
    const size_t GSUME_OFF = 10304;                   // 64
    const size_t GSUM2_OFF = 10368;                   // 64
    const size_t BPART_OFF = 11264;                   // 64*256 (reused 3x)
    const size_t SIM_OFF   = 32768;                   // 64*65536 (sim -> e)
    const size_t WP_OFF    = SIM_OFF + (size_t)NB * NN;   // 64*65536
    const size_t RPART_OFF = WP_OFF + (size_t)NB * NN;    // 64*256*64

    float* w_out = out + NB * MD;   // w section of d_out

    ntm_gemm_wmma<<<20, 32, 0, stream>>>(emb, W, bias, ws + A_OFF);
    ntm_prep<<<NB, 64, 0, stream>>>(ws + A_OFF, ws + KN_OFF, ws + SC_OFF);
    ntm_sim<<<dim3(CHUNKS, NB), 256, 0, stream>>>(mem, ws + KN_OFF,
                                                  ws + SIM_OFF, ws + BPART_OFF);
    ntm_reduce_max<<<NB, 256, 0, stream>>>(ws + BPART_OFF, ws + GMAX_OFF);
    ntm_exp<<<dim3(CHUNKS, NB), 256, 0, stream>>>(ws + SIM_OFF, ws + GMAX_OFF,
                                                  ws + SC_OFF, ws + BPART_OFF);
    ntm_reduce_sum<<<NB, 256, 0, stream>>>(ws + BPART_OFF, ws + GSUME_OFF);
    ntm_shift_pow<<<dim3(CHUNKS, NB), 256, 0, stream>>>(ws + SIM_OFF, wprev,
                                                        ws + SC_OFF, ws + GSUME_OFF,
                                                        ws + WP_OFF, ws + BPART_OFF);
    ntm_reduce_sum<<<NB, 256, 0, stream>>>(ws + BPART_OFF, ws + GSUM2_OFF);
    ntm_norm_w<<<dim3(CHUNKS, NB), 256, 0, stream>>>(ws + WP_OFF, ws + GSUM2_OFF, w_out);
    ntm_read<<<dim3(CHUNKS, NB), 256, 0, stream>>>(mem, w_out, ws + RPART_OFF);
    ntm_read_reduce<<<NB, 64, 0, stream>>>(ws + RPART_OFF, out);
}